// Set2Set_17875653886191
// MI455X (gfx1250) — compile-verified
//
#include <hip/hip_runtime.h>
#include <math.h>

// ---------------------------------------------------------------------------
// Set2Set (6 iterations) for MI455X / gfx1250, wave32.
//   D = 256 (hidden), B = 1024 graphs, N = 200k nodes.
// Per iteration:
//   K1: LSTM cell as fp32 WMMA GEMM (gates never hit memory), fused pointwise.
//   K2: per-graph attention (segment softmax + weighted readout), no atomics
//       across graphs because segment_ids are sorted (contiguous ranges).
// Roofline: ~11 GFLOP total vs ~2.5 GB of feat traffic -> memory bound on the
// attention passes; LSTM GEMM is tiny and precision-preserving via f32 WMMA.
// ---------------------------------------------------------------------------

typedef float v2f __attribute__((ext_vector_type(2)));
typedef float v8f __attribute__((ext_vector_type(8)));

#define DH    256          // hidden size D
#define TWOD  512
#define FOURD 1024

__device__ __forceinline__ float sigf(float x) {
    return 1.0f / (1.0f + __expf(-x));
}
__device__ __forceinline__ float tanh_fast(float x) {
    // robust at +/- inf: 1 - 2/(e^{2x}+1)
    float e2 = __expf(2.0f * x);
    return 1.0f - 2.0f / (e2 + 1.0f);
}

// ---------------------------------------------------------------------------
// Init: zero q_star (buffer 0 = d_out) and c; compute segment start offsets
// by binary search over the sorted segment_ids.
// ---------------------------------------------------------------------------
__global__ void init_kernel(const int* __restrict__ seg_ids, int N, int B,
                            float* __restrict__ qstar0,   // [B, 2D]
                            float* __restrict__ cbuf,     // [B, D]
                            int* __restrict__ seg_starts) // [B+1]
{
    int i = blockIdx.x * blockDim.x + threadIdx.x;
    int BD = B * DH;
    if (i < BD)       cbuf[i]   = 0.0f;
    if (i < 2 * BD)   qstar0[i] = 0.0f;
    if (i <= B) {
        int lo = 0, hi = N;               // lower_bound(seg_ids, i)
        while (lo < hi) {
            int mid = (lo + hi) >> 1;
            if (seg_ids[mid] < i) lo = mid + 1; else hi = mid;
        }
        seg_starts[i] = lo;
    }
}

// ---------------------------------------------------------------------------
// K1: LSTM cell via V_WMMA_F32_16X16X4_F32.
//   gates = [q_prev | q_prev[:, :D]] @ [W_ih | W_hh]^T + (b_ih + b_hh)
//   c = sig(gf)*c + sig(gi)*tanh(gg);  h = sig(go)*tanh(c)
//   writes h into q_next[:, :D];  c updated in place (unique owner per elem).
// One wave per (16-row tile, 16-col tile within D); computes all 4 gate tiles
// so the A fragment is reused 4x per K-step.
//
// fp32 WMMA fragment layout (ISA 7.12.2):
//   A 16x4 : lane l -> row m=l&15, k = 2*(l>>4) + {vgpr0,vgpr1}
//   B 4x16 : lane l -> col n=l&15, k = 2*(l>>4) + {vgpr0,vgpr1}   (B[k,n]=W[n,k])
//   C 16x16: lane l -> col n=l&15, rows r + 8*(l>>4), r = vgpr 0..7
// ---------------------------------------------------------------------------
__global__ __launch_bounds__(32)
void lstm_wmma_kernel(const float* __restrict__ q_prev,  // [B, 2D]
                      const float* __restrict__ W_ih,    // [4D, 2D] row-major
                      const float* __restrict__ W_hh,    // [4D, D]  row-major
                      const float* __restrict__ b_ih,    // [4D]
                      const float* __restrict__ b_hh,    // [4D]
                      float* __restrict__ cbuf,          // [B, D] in-place
                      float* __restrict__ q_next)        // [B, 2D]: write [:, :D]
{
    const int lane  = threadIdx.x;        // 0..31
    const int jt    = blockIdx.x;         // col tile within D: 0..D/16-1
    const int rt    = blockIdx.y;         // row tile: 0..B/16-1
    const int mlo   = lane & 15;
    const int hi    = lane >> 4;          // 0 or 1
    const int koff  = 2 * hi;
    const int rbase = rt * 16;
    const int nloc  = jt * 16 + mlo;      // column within D for B/C/D frags

    v8f acc[4];
    const v8f vzero = {0.f, 0.f, 0.f, 0.f, 0.f, 0.f, 0.f, 0.f};
#pragma unroll
    for (int g = 0; g < 4; ++g) acc[g] = vzero;

    const float* Arow = q_prev + (size_t)(rbase + mlo) * TWOD;

    const float* Wih_rows[4];
    const float* Whh_rows[4];
#pragma unroll
    for (int g = 0; g < 4; ++g) {
        int n = g * DH + nloc;            // gate row in W (0..4D)
        Wih_rows[g] = W_ih + (size_t)n * TWOD;
        Whh_rows[g] = W_hh + (size_t)n * DH;
    }

    // ---- K part 1: A = q_prev (K = 0..511), W_ih --------------------------
    for (int kk = 0; kk < TWOD; kk += 4) {
        v2f a = *(const v2f*)(Arow + kk + koff);
#pragma unroll
        for (int g = 0; g < 4; ++g) {
            v2f b = *(const v2f*)(Wih_rows[g] + kk + koff);
            acc[g] = __builtin_amdgcn_wmma_f32_16x16x4_f32(
                false, a, false, b, (short)0, acc[g], false, false);
        }
    }
    // ---- K part 2: A = h_prev = q_prev[:, :D] (K = 0..255), W_hh ----------
    for (int kk = 0; kk < DH; kk += 4) {
        v2f a = *(const v2f*)(Arow + kk + koff);
#pragma unroll
        for (int g = 0; g < 4; ++g) {
            v2f b = *(const v2f*)(Whh_rows[g] + kk + koff);
            acc[g] = __builtin_amdgcn_wmma_f32_16x16x4_f32(
                false, a, false, b, (short)0, acc[g], false, false);
        }
    }

    float bias[4];
#pragma unroll
    for (int g = 0; g < 4; ++g) {
        int n = g * DH + nloc;
        bias[g] = b_ih[n] + b_hh[n];
    }

    // fused pointwise LSTM update (each (row, col) owned by exactly one lane/r)
#pragma unroll
    for (int r = 0; r < 8; ++r) {
        int mg = rbase + r + 8 * hi;           // graph row
        size_t idx = (size_t)mg * DH + nloc;
        float gi = acc[0][r] + bias[0];
        float gf = acc[1][r] + bias[1];
        float gg = acc[2][r] + bias[2];
        float go = acc[3][r] + bias[3];
        float cn = sigf(gf) * cbuf[idx] + sigf(gi) * tanh_fast(gg);
        float hn = sigf(go) * tanh_fast(cn);
        cbuf[idx] = cn;
        q_next[(size_t)mg * TWOD + nloc] = hn;   // h == q == q_star[:, :D]
    }
}

// ---------------------------------------------------------------------------
// K2: per-graph attention + softmax + readout. One 256-thread block per graph.
//   Phase 1: e[n] = <feat[n], q>, wave-per-node, running max.
//   Phase 2: ex = exp(e - max), block sum (LDS atomic, ds_add_f32).
//   Phase 3: readout[d] = sum_n feat[n,d]*ex[n] / denom, thread-per-feature,
//            alpha staged through LDS in chunks of 256 nodes.
// Writes q_next[:, D:2D].
// ---------------------------------------------------------------------------
__global__ __launch_bounds__(256)
void attention_kernel(const float* __restrict__ feat,      // [N, D]
                      const int*   __restrict__ seg_starts,// [B+1]
                      float* __restrict__ e_buf,           // [N] scratch
                      float* __restrict__ q_next)          // [B, 2D]
{
    const int b    = blockIdx.x;
    const int tid  = threadIdx.x;
    const int lane = tid & 31;
    const int w    = tid >> 5;            // wave id 0..7

    __shared__ float q_lds[DH];
    __shared__ float s_alpha[256];
    __shared__ float wmax[8];
    __shared__ float s_max;
    __shared__ float s_sum;

    const int s = seg_starts[b];
    const int t = seg_starts[b + 1];

    q_lds[tid] = q_next[(size_t)b * TWOD + tid];   // q = h just written by K1
    if (tid == 0) s_sum = 0.0f;
    __syncthreads();

    // ---- Phase 1: e[n] and per-wave max -----------------------------------
    float lmax = -INFINITY;
    for (int n = s + w; n < t; n += 8) {
        const float* fr = feat + (size_t)n * DH;
        float p = 0.0f;
#pragma unroll
        for (int j = 0; j < 8; ++j) {
            int d = lane + 32 * j;                 // coalesced 128B per j
            p += fr[d] * q_lds[d];
        }
#pragma unroll
        for (int off = 16; off > 0; off >>= 1)
            p += __shfl_xor(p, off, 32);
        if (lane == 0) e_buf[n] = p;
        lmax = fmaxf(lmax, p);                     // identical across lanes
    }
    if (lane == 0) wmax[w] = lmax;
    __syncthreads();
    if (tid == 0) {
        float m = wmax[0];
#pragma unroll
        for (int i = 1; i < 8; ++i) m = fmaxf(m, wmax[i]);
        s_max = m;
    }
    __syncthreads();

    // ---- Phase 2: ex = exp(e - max); denom --------------------------------
    const float gmax = s_max;
    float lsum = 0.0f;
    for (int n = s + tid; n < t; n += 256) {
        float ex = __expf(e_buf[n] - gmax);
        e_buf[n] = ex;
        lsum += ex;
    }
    atomicAdd(&s_sum, lsum);                       // ds_add_f32
    __syncthreads();
    const float denom = s_sum;

    // ---- Phase 3: readout[d], thread-per-feature --------------------------
    float acc = 0.0f;
    for (int base = s; base < t; base += 256) {
        int cnt = min(256, t - base);
        __syncthreads();
        if (tid < cnt) s_alpha[tid] = e_buf[base + tid];
        __syncthreads();
#pragma unroll 4
        for (int j = 0; j < cnt; ++j)
            acc += feat[(size_t)(base + j) * DH + tid] * s_alpha[j];
    }
    float ro = (denom > 0.0f) ? (acc / denom) : 0.0f;  // empty-segment guard
    q_next[(size_t)b * TWOD + DH + tid] = ro;
}

// ---------------------------------------------------------------------------
// Host launcher. Inputs: feat, W_ih, W_hh, b_ih, b_hh, segment_ids, batch_size
// Output: q_star [B, 2D] fp32.
// ---------------------------------------------------------------------------
extern "C" void kernel_launch(void* const* d_in, const int* in_sizes, int n_in,
                              void* d_out, int out_size, void* d_ws, size_t ws_size,
                              hipStream_t stream) {
    const float* feat = (const float*)d_in[0];
    const float* W_ih = (const float*)d_in[1];
    const float* W_hh = (const float*)d_in[2];
    const float* b_ih = (const float*)d_in[3];
    const float* b_hh = (const float*)d_in[4];
    const int*   seg  = (const int*)d_in[5];

    const int N = in_sizes[0] / DH;      // 200000
    const int B = out_size / TWOD;       // 1024

    // workspace layout: e[N] | qstar_alt[B*2D] | seg_starts[B+1] | c[B*D]
    float* e_buf     = (float*)d_ws;
    float* qstar_alt = e_buf + N;
    int*   seg_st    = (int*)(qstar_alt + (size_t)B * TWOD);
    float* cbuf      = (float*)(seg_st + (B + 1));

    float* qbuf[2] = { (float*)d_out, qstar_alt };   // final lands in d_out

    {
        int total  = 2 * B * DH;                     // covers q, c, and B+1
        int blocks = (total + 255) / 256;
        init_kernel<<<blocks, 256, 0, stream>>>(seg, N, B, qbuf[0], cbuf, seg_st);
    }

    dim3 lstm_grid(DH / 16, B / 16);                 // (16, 64) waves

    for (int it = 0; it < 6; ++it) {
        const float* q_prev = qbuf[it & 1];
        float*       q_next = qbuf[(it + 1) & 1];
        lstm_wmma_kernel<<<lstm_grid, 32, 0, stream>>>(
            q_prev, W_ih, W_hh, b_ih, b_hh, cbuf, q_next);
        attention_kernel<<<B, 256, 0, stream>>>(feat, seg_st, e_buf, q_next);
    }
}